// SEBlock_75883482185780
// MI455X (gfx1250) — compile-verified
//
#include <hip/hip_runtime.h>

// SE block: x[32,256,112,112] f32
//   s = mean_{H,W}(x)                       [32,256]
//   h = relu(s @ w1^T + b1)                 [32,16]
//   scale = sigmoid(h @ w2^T + b2)          [32,256]
//   out = x * scale[:, :, None, None]
//
// Memory-bound: ~1.23 GB HBM traffic -> ~53us at 23.3 TB/s.
// Tiny FCs done with V_WMMA_F32_16X16X4_F32 in a single wave32 workgroup.

typedef __attribute__((ext_vector_type(2))) float v2f;
typedef __attribute__((ext_vector_type(4))) float v4f;
typedef __attribute__((ext_vector_type(8))) float v8f;

#define SE_B   32
#define SE_C   256
#define SE_H   16
#define SE_HW  12544          // 112*112
#define SE_HW4 3136           // HW/4

// ---------------------------------------------------------------------------
// Kernel 1: global average pool. One block per (b,c) plane (8192 blocks).
// Streams 411 MB with nontemporal b128 loads; wave32 shuffle + LDS reduce.
// ---------------------------------------------------------------------------
__global__ __launch_bounds__(256) void se_pool_kernel(const float* __restrict__ x,
                                                      float* __restrict__ s_g) {
    const int plane = blockIdx.x;                        // b*C + c
    const v4f* __restrict__ xp =
        (const v4f*)(x + (size_t)plane * SE_HW);

    float sum = 0.0f;
    for (int i = threadIdx.x; i < SE_HW4; i += 256) {
        v4f v = __builtin_nontemporal_load(&xp[i]);      // global_load_b128, TH=NT
        sum += (v.x + v.y) + (v.z + v.w);
    }
    // wave32 reduction
    #pragma unroll
    for (int off = 16; off > 0; off >>= 1)
        sum += __shfl_down(sum, off, 32);

    __shared__ float partial[8];
    const int wave = threadIdx.x >> 5;
    if ((threadIdx.x & 31) == 0) partial[wave] = sum;
    __syncthreads();
    if (threadIdx.x == 0) {
        float tot = 0.0f;
        #pragma unroll
        for (int w = 0; w < 8; ++w) tot += partial[w];
        s_g[plane] = tot * (1.0f / (float)SE_HW);
    }
}

// ---------------------------------------------------------------------------
// Kernel 2: excite (two tiny GEMMs) on one wave32 using V_WMMA_F32_16X16X4_F32.
//
// ISA operand striping (CDNA5 ISA, 32-bit 16x4 A / 4x16 B / 16x16 C):
//   A: lane L (row M = L&15): VGPR0/1 hold K = 2*(L>>4) + {0,1}
//   B: lane L (col N = L&15): VGPR0/1 hold K = 2*(L>>4) + {0,1}
//   C/D: VGPR r, lane L: element (M = r + 8*(L>>4), N = L&15)
// ---------------------------------------------------------------------------
__global__ __launch_bounds__(32) void se_fc_kernel(const float* __restrict__ s_g,
                                                   const float* __restrict__ w1,  // [16,256]
                                                   const float* __restrict__ b1,  // [16]
                                                   const float* __restrict__ w2,  // [256,16]
                                                   const float* __restrict__ b2,  // [256]
                                                   float* __restrict__ scale_g) { // [32,256]
    __shared__ float h_lds[SE_B * SE_H];                 // 2 KB

    const int tid  = threadIdx.x;                        // 0..31, EXEC all-1s
    const int lane = tid & 15;
    const int half = tid >> 4;                           // selects K pair

    // ---- GEMM1: h = relu(s @ w1^T + b1), M=32 N=16 K=256 ----
    #pragma unroll
    for (int t = 0; t < 2; ++t) {                        // M tiles
        v8f acc = {};
        const float* __restrict__ arow = s_g + (size_t)(t * 16 + lane) * SE_C + 2 * half;
        const float* __restrict__ brow = w1  + (size_t)lane * SE_C + 2 * half;
        for (int k0 = 0; k0 < SE_C; k0 += 4) {
            v2f a, b;
            a[0] = arow[k0];     a[1] = arow[k0 + 1];
            b[0] = brow[k0];     b[1] = brow[k0 + 1];
            acc = __builtin_amdgcn_wmma_f32_16x16x4_f32(
                false, a, false, b, (short)0, acc, false, false);
        }
        const float bias = b1[lane];
        #pragma unroll
        for (int r = 0; r < 8; ++r) {
            const int m = t * 16 + r + 8 * half;
            const float v = acc[r] + bias;
            h_lds[m * SE_H + lane] = v > 0.0f ? v : 0.0f;
        }
    }
    __syncthreads();

    // ---- GEMM2: scale = sigmoid(h @ w2^T + b2), M=32 N=256 K=16 ----
    for (int t = 0; t < 2; ++t) {                        // M tiles
        for (int j = 0; j < 16; ++j) {                   // N tiles of 16 channels
            const int c0 = j * 16;
            v8f acc = {};
            const float* __restrict__ arow = h_lds + (t * 16 + lane) * SE_H + 2 * half;
            const float* __restrict__ brow = w2 + (size_t)(c0 + lane) * SE_H + 2 * half;
            #pragma unroll
            for (int k0 = 0; k0 < SE_H; k0 += 4) {
                v2f a, b;
                a[0] = arow[k0];  a[1] = arow[k0 + 1];
                b[0] = brow[k0];  b[1] = brow[k0 + 1];
                acc = __builtin_amdgcn_wmma_f32_16x16x4_f32(
                    false, a, false, b, (short)0, acc, false, false);
            }
            const float bias = b2[c0 + lane];
            #pragma unroll
            for (int r = 0; r < 8; ++r) {
                const int m = t * 16 + r + 8 * half;
                const float v = acc[r] + bias;
                scale_g[m * SE_C + c0 + lane] = 1.0f / (1.0f + __expf(-v));
            }
        }
    }
}

// ---------------------------------------------------------------------------
// Kernel 3: out = x * scale. One block per (plane, half-plane): scale is
// block-uniform (scalar), streams 822 MB with nontemporal b128 load/store.
// ---------------------------------------------------------------------------
__global__ __launch_bounds__(256) void se_scale_kernel(const float* __restrict__ x,
                                                       const float* __restrict__ sc,
                                                       float* __restrict__ out) {
    const int plane = blockIdx.x;                        // b*C + c
    const float s   = sc[plane];                         // uniform per block
    const size_t base4 = (size_t)plane * SE_HW4 + (size_t)blockIdx.y * (SE_HW4 / 2);

    const v4f* __restrict__ xp = (const v4f*)x + base4;
    v4f* __restrict__ op       = (v4f*)out + base4;

    for (int i = threadIdx.x; i < SE_HW4 / 2; i += 256) {
        v4f v = __builtin_nontemporal_load(&xp[i]);
        v *= s;
        __builtin_nontemporal_store(v, &op[i]);
    }
}

// ---------------------------------------------------------------------------
extern "C" void kernel_launch(void* const* d_in, const int* in_sizes, int n_in,
                              void* d_out, int out_size, void* d_ws, size_t ws_size,
                              hipStream_t stream) {
    const float* x  = (const float*)d_in[0];   // [32,256,112,112]
    const float* w1 = (const float*)d_in[1];   // [16,256]
    const float* b1 = (const float*)d_in[2];   // [16]
    const float* w2 = (const float*)d_in[3];   // [256,16]
    const float* b2 = (const float*)d_in[4];   // [256]
    float* out = (float*)d_out;

    float* s_g     = (float*)d_ws;             // [32*256] squeeze result
    float* scale_g = s_g + SE_B * SE_C;        // [32*256] sigmoid scales

    se_pool_kernel<<<dim3(SE_B * SE_C), dim3(256), 0, stream>>>(x, s_g);
    se_fc_kernel<<<dim3(1), dim3(32), 0, stream>>>(s_g, w1, b1, w2, b2, scale_g);
    se_scale_kernel<<<dim3(SE_B * SE_C, 2), dim3(256), 0, stream>>>(x, scale_g, out);
}